// Model_26388279067451
// MI455X (gfx1250) — compile-verified
//
#include <hip/hip_runtime.h>

typedef __attribute__((ext_vector_type(2))) float v2f;
typedef __attribute__((ext_vector_type(4))) float v4f;
typedef __attribute__((ext_vector_type(8))) float v8f;
typedef int v4i_vs __attribute__((vector_size(4 * sizeof(int))));

#if __has_builtin(__builtin_amdgcn_global_load_async_to_lds_b128)
#define HAVE_ASYNC_COPY 1
#else
#define HAVE_ASYNC_COPY 0
#endif

#if __has_builtin(__builtin_amdgcn_s_wait_asynccnt)
#define WAIT_ASYNC0() __builtin_amdgcn_s_wait_asynccnt(0)
#else
#define WAIT_ASYNC0() asm volatile("s_wait_asynccnt 0" ::: "memory")
#endif

static constexpr int Ld = 8192;
static constexpr int Bd = 4096;
static constexpr int Dd = 1024;

static constexpr int BM = 128;   // M tile (pred rows)
static constexpr int BN = 64;    // N tile (gt rows)
static constexpr int BK = 32;    // K tile (8 chunks of 4)

#if HAVE_ASYNC_COPY
static __device__ __forceinline__ void async_copy_b128(const v4f* g, v4f* lds) {
    // global -> LDS DMA, tracked by ASYNCcnt (emits global_load_async_to_lds_b128)
    __builtin_amdgcn_global_load_async_to_lds_b128(
        (v4i_vs*)g, (v4i_vs*)lds, 0, 0);
}
#endif

// -------------------- zero the accumulators (must run every call) ------------
__global__ void k_init(float* acc) {
    if (threadIdx.x == 0) { acc[0] = 0.0f; acc[1] = 0.0f; }
}

// -------------------- local loss: sum_i <q_hat_i, q_real_i> ------------------
__global__ __launch_bounds__(256) void k_localdot(const float* __restrict__ qh,
                                                  const float* __restrict__ qr,
                                                  float* __restrict__ acc) {
    const int n4 = (Ld * Dd) / 4;
    float s = 0.0f;
    for (int i = blockIdx.x * blockDim.x + threadIdx.x; i < n4;
         i += gridDim.x * blockDim.x) {
        v4f a = ((const v4f*)qh)[i];
        v4f b = ((const v4f*)qr)[i];
        s += a[0] * b[0] + a[1] * b[1] + a[2] * b[2] + a[3] * b[3];
    }
    #pragma unroll
    for (int off = 16; off; off >>= 1) s += __shfl_xor(s, off, 32);
    if ((threadIdx.x & 31) == 0) atomicAdd(acc, s);
}

// -------------------- diagonal: pos[i] = <pred_i, gt_i> ----------------------
__global__ __launch_bounds__(256) void k_diag(const float* __restrict__ pred,
                                              const float* __restrict__ gt,
                                              float* __restrict__ pos) {
    const int wave = threadIdx.x >> 5;
    const int lane = threadIdx.x & 31;
    const int row  = blockIdx.x * 8 + wave;
    const float* p = pred + (size_t)row * Dd;
    const float* g = gt   + (size_t)row * Dd;
    float s = 0.0f;
    #pragma unroll 4
    for (int k = lane; k < Dd; k += 32) s += p[k] * g[k];
    #pragma unroll
    for (int off = 16; off; off >>= 1) s += __shfl_xor(s, off, 32);
    if (lane == 0) pos[row] = s;
}

// ------------- fused GEMM (fp32 WMMA 16x16x4) + hinge reduction --------------
__global__ __launch_bounds__(256) void k_gemm_hinge(
        const float* __restrict__ pred,   // [B, D]  (M side)
        const float* __restrict__ gt,     // [B, D]  (N side)
        const float* __restrict__ pos,    // [B] diagonal scores
        float* __restrict__ hacc) {
    // Double-buffered LDS, staged as float4 K-chunks: As[b][c*BM+m] = row m, k=4c..4c+3
    __shared__ v4f As[2][(BK / 4) * BM];  // 2 x 16 KB
    __shared__ v4f Bs[2][(BK / 4) * BN];  // 2 x  8 KB

    const int tid  = threadIdx.x;
    const int lane = tid & 31;
    const int wave = tid >> 5;
    const int h    = lane >> 4;          // half: selects K pair {0,1} vs {2,3}
    const int ml   = lane & 15;          // row (A) / col (B) within 16x16
    const int wm   = wave & 3;           // wave M group: 4 x 32 rows
    const int wn   = wave >> 2;          // wave N group: 2 x 32 cols

    const int m0 = blockIdx.y * BM;
    const int n0 = blockIdx.x * BN;

    // per-thread staging assignments (fixed across K tiles)
    const v4f* gA[4]; int dA[4];
    #pragma unroll
    for (int i = 0; i < 4; ++i) {
        int idx = tid + i * 256;
        int c = idx & 7, r = idx >> 3;          // 128 rows x 8 chunks
        gA[i] = (const v4f*)(pred + (size_t)(m0 + r) * Dd) + c;
        dA[i] = c * BM + r;
    }
    const v4f* gB[2]; int dB[2];
    #pragma unroll
    for (int i = 0; i < 2; ++i) {
        int idx = tid + i * 256;
        int c = idx & 7, r = idx >> 3;          // 64 rows x 8 chunks
        gB[i] = (const v4f*)(gt + (size_t)(n0 + r) * Dd) + c;
        dB[i] = c * BN + r;
    }

    v8f acc[2][2] = {};

    auto compute = [&](const v4f* Ab, const v4f* Bb) {
        #pragma unroll
        for (int c = 0; c < 8; ++c) {
            v2f a[2], b[2];
            #pragma unroll
            for (int i = 0; i < 2; ++i) {
                const v2f* p = (const v2f*)&Ab[c * BM + wm * 32 + i * 16 + ml];
                a[i] = p[h];   // conflict-free: 64 distinct banks across wave
            }
            #pragma unroll
            for (int j = 0; j < 2; ++j) {
                const v2f* p = (const v2f*)&Bb[c * BN + wn * 32 + j * 16 + ml];
                b[j] = p[h];
            }
            #pragma unroll
            for (int i = 0; i < 2; ++i)
                #pragma unroll
                for (int j = 0; j < 2; ++j)
                    acc[i][j] = __builtin_amdgcn_wmma_f32_16x16x4_f32(
                        false, a[i], false, b[j], (short)0, acc[i][j],
                        false, false);
        }
    };

    constexpr int NT = Dd / BK;   // 32 K-tiles, chunk stride = BK/4 = 8 float4

#if HAVE_ASYNC_COPY
    // ---- pipeline: async DMA tile t+1 while WMMAs chew on tile t ----
    #pragma unroll
    for (int i = 0; i < 4; ++i) async_copy_b128(gA[i], &As[0][dA[i]]);
    #pragma unroll
    for (int i = 0; i < 2; ++i) async_copy_b128(gB[i], &Bs[0][dB[i]]);
    WAIT_ASYNC0();
    __syncthreads();
    for (int t = 0; t < NT; ++t) {
        const int cur = t & 1;
        if (t + 1 < NT) {
            const int co = (t + 1) * (BK / 4);
            #pragma unroll
            for (int i = 0; i < 4; ++i)
                async_copy_b128(gA[i] + co, &As[cur ^ 1][dA[i]]);
            #pragma unroll
            for (int i = 0; i < 2; ++i)
                async_copy_b128(gB[i] + co, &Bs[cur ^ 1][dB[i]]);
        }
        compute(As[cur], Bs[cur]);
        if (t + 1 < NT) { WAIT_ASYNC0(); __syncthreads(); }
    }
#else
    // ---- fallback: register double-buffer (batched loads, one wait) ----
    v4f ra[4], rb[2];
    #pragma unroll
    for (int i = 0; i < 4; ++i) ra[i] = gA[i][0];
    #pragma unroll
    for (int i = 0; i < 2; ++i) rb[i] = gB[i][0];
    for (int t = 0; t < NT; ++t) {
        const int cur = t & 1;
        #pragma unroll
        for (int i = 0; i < 4; ++i) As[cur][dA[i]] = ra[i];
        #pragma unroll
        for (int i = 0; i < 2; ++i) Bs[cur][dB[i]] = rb[i];
        if (t + 1 < NT) {
            const int co = (t + 1) * (BK / 4);
            #pragma unroll
            for (int i = 0; i < 4; ++i) ra[i] = gA[i][co];
            #pragma unroll
            for (int i = 0; i < 2; ++i) rb[i] = gB[i][co];
        }
        __syncthreads();
        compute(As[cur], Bs[cur]);
    }
#endif

    // ---- hinge epilogue: VGPR r of C maps to M = r + 8*h (lanes split) ----
    float posv[2][8];
    #pragma unroll
    for (int i = 0; i < 2; ++i)
        #pragma unroll
        for (int r = 0; r < 8; ++r)
            posv[i][r] = pos[m0 + wm * 32 + i * 16 + 8 * h + r];

    float s = 0.0f;
    #pragma unroll
    for (int i = 0; i < 2; ++i)
        #pragma unroll
        for (int j = 0; j < 2; ++j)
            #pragma unroll
            for (int r = 0; r < 8; ++r) {
                float v = acc[i][j][r] - posv[i][r] + 1.0f;
                s += (v > 0.0f) ? v : 0.0f;
            }

    #pragma unroll
    for (int off = 16; off; off >>= 1) s += __shfl_xor(s, off, 32);
    if (lane == 0) atomicAdd(hacc, s);
}

// -------------------- finalize: out = -local_sum/L + hinge_sum ---------------
__global__ void k_final(const float* __restrict__ acc, float* __restrict__ out) {
    if (threadIdx.x == 0) out[0] = -(acc[0] / (float)Ld) + acc[1];
}

extern "C" void kernel_launch(void* const* d_in, const int* in_sizes, int n_in,
                              void* d_out, int out_size, void* d_ws, size_t ws_size,
                              hipStream_t stream) {
    const float* q_hat  = (const float*)d_in[0];   // [L, D]
    const float* q_real = (const float*)d_in[1];   // [L, D]
    const float* gt     = (const float*)d_in[2];   // encoded_gt   [B, D]
    const float* pred   = (const float*)d_in[3];   // encoded_pred [B, D]

    float* wsf = (float*)d_ws;     // [0]=local acc, [1]=hinge acc, [2..2+B)=pos
    float* pos = wsf + 2;

    k_init<<<1, 64, 0, stream>>>(wsf);
    k_localdot<<<1024, 256, 0, stream>>>(q_hat, q_real, wsf);
    k_diag<<<Bd / 8, 256, 0, stream>>>(pred, gt, pos);

    dim3 grid(Bd / BN, Bd / BM);   // (64, 32)
    k_gemm_hinge<<<grid, 256, 0, stream>>>(pred, gt, pos, wsf + 1);

    k_final<<<1, 1, 0, stream>>>(wsf, (float*)d_out);
}